// SelectableDense_20504173871256
// MI455X (gfx1250) — compile-verified
//
#include <hip/hip_runtime.h>
#include <hip/hip_bf16.h>
#include <stdint.h>

typedef __attribute__((ext_vector_type(2))) float v2f;
typedef __attribute__((ext_vector_type(8))) float v8f;

#define NUM_OPTIONS 64
#define IN_FEAT 512
#define OUT_FEAT 512
#define BATCH 2048

// ---------------------------------------------------------------------------
// Kernel 1: bucket samples by option index.
// ws layout: int offsets[65] | int ids[BATCH]
// ---------------------------------------------------------------------------
__global__ __launch_bounds__(256) void sd_bucket_kernel(const int* __restrict__ indices,
                                                        int* __restrict__ offsets,
                                                        int* __restrict__ ids) {
    __shared__ int cnt[NUM_OPTIONS];
    __shared__ int off[NUM_OPTIONS + 1];
    __shared__ int cur[NUM_OPTIONS];
    const int t = threadIdx.x;

    if (t < NUM_OPTIONS) cnt[t] = 0;
    __syncthreads();

    for (int b = t; b < BATCH; b += blockDim.x) {
        atomicAdd(&cnt[indices[b] & (NUM_OPTIONS - 1)], 1);
    }
    __syncthreads();

    if (t == 0) {
        int s = 0;
        for (int o = 0; o < NUM_OPTIONS; ++o) { off[o] = s; s += cnt[o]; }
        off[NUM_OPTIONS] = s;
    }
    __syncthreads();

    if (t < NUM_OPTIONS) cur[t] = off[t];
    if (t < NUM_OPTIONS + 1) offsets[t] = off[t];
    __syncthreads();

    for (int b = t; b < BATCH; b += blockDim.x) {
        int p = atomicAdd(&cur[indices[b] & (NUM_OPTIONS - 1)], 1);
        ids[p] = b;
    }
}

// ---------------------------------------------------------------------------
// Kernel 2: per-option GEMM using V_WMMA_F32_16X16X4_F32 (exact f32 FMA path).
// grid.x = NUM_OPTIONS * 4 (4 column-blocks of 128 cols); block = 256 (8 waves).
// Wave w handles N strip [nblk*128 + w*16, +16). Loops over 16-row m-tiles of
// this option's bucket; X tile staged in LDS (padded row stride vs bank
// conflicts); per-tile output row bases staged in LDS so the store epilogue
// uses ds_load instead of serialized dependent global loads.
// ---------------------------------------------------------------------------
#define XS_STRIDE 516  // 512 + 4 pad: lane-stride across rows maps to distinct LDS banks

__global__ __launch_bounds__(256) void sd_gemm_kernel(const float* __restrict__ x,
                                                      const float* __restrict__ w,
                                                      const float* __restrict__ bias,
                                                      const int* __restrict__ offsets,
                                                      const int* __restrict__ ids,
                                                      float* __restrict__ out) {
    __shared__ float Xs[16 * XS_STRIDE];  // 33,024 B
    __shared__ int   rowbase[16];         // ids[mt+m] * OUT_FEAT (0 for pad rows)

    const int o    = blockIdx.x >> 2;          // option id
    const int nblk = blockIdx.x & 3;           // which 128-col block
    const int wave = threadIdx.x >> 5;         // 0..7
    const int lane = threadIdx.x & 31;
    const int nbase = nblk * 128 + wave * 16;  // this wave's N strip

    const int begin = offsets[o];
    const int end   = offsets[o + 1];

    const float* __restrict__ Wo = w + (size_t)o * OUT_FEAT * IN_FEAT;

    const int kh   = lane >> 4;    // 0 or 1: which K-half of the fragment
    const int mrow = lane & 15;    // A-fragment row for this lane
    const int ncol = lane & 15;    // B/D-fragment column for this lane

    // B fragment source row: W[nbase + ncol][*]
    const float* __restrict__ wrow = Wo + (size_t)(nbase + ncol) * IN_FEAT;
    const float bv = bias[(size_t)o * OUT_FEAT + nbase + ncol];

    for (int mt = begin; mt < end; mt += 16) {
        const int rows = (end - mt < 16) ? (end - mt) : 16;

        // ---- stage per-row output bases (read again cheaply from LDS in epilogue)
        if (threadIdx.x < 16) {
            rowbase[threadIdx.x] =
                (threadIdx.x < rows) ? ids[mt + threadIdx.x] * OUT_FEAT : 0;
        }

        // ---- stage gathered X rows (16 x 512 f32) into LDS, zero-fill tail ----
        for (int i = threadIdx.x; i < 16 * (IN_FEAT / 4); i += 256) {
            const int m  = i >> 7;        // row 0..15
            const int c4 = i & 127;       // float4 column
            float4 v = make_float4(0.f, 0.f, 0.f, 0.f);
            if (m < rows) {
                v = *(const float4*)(x + (size_t)ids[mt + m] * IN_FEAT + c4 * 4);
            }
            *(float4*)(&Xs[m * XS_STRIDE + c4 * 4]) = v;
        }
        __syncthreads();

        // ---- K loop: 128 x v_wmma_f32_16x16x4_f32 ----
        v8f acc = {};
        const float* __restrict__ arow = &Xs[mrow * XS_STRIDE];
        #pragma unroll 8
        for (int ks = 0; ks < IN_FEAT; ks += 4) {
            v2f a, b;
            a.x = arow[ks + 2 * kh + 0];
            a.y = arow[ks + 2 * kh + 1];
            b.x = wrow[ks + 2 * kh + 0];
            b.y = wrow[ks + 2 * kh + 1];
            // 8 args: (neg_a, A, neg_b, B, c_mod, C, reuse_a, reuse_b)
            acc = __builtin_amdgcn_wmma_f32_16x16x4_f32(
                false, a, false, b, (short)0, acc, false, false);
        }

        // ---- store D tile + bias; VGPR r, lanes[16h..] -> (m = r + 8h, n) ----
        const int mhalf = kh * 8;
        #pragma unroll
        for (int r = 0; r < 8; ++r) {
            const int m = r + mhalf;
            if (m < rows) {
                out[(size_t)(rowbase[m] + nbase + ncol)] = acc[r] + bv;
            }
        }
        __syncthreads();  // Xs/rowbase consumed before next tile overwrites them
    }
}

// ---------------------------------------------------------------------------
extern "C" void kernel_launch(void* const* d_in, const int* in_sizes, int n_in,
                              void* d_out, int out_size, void* d_ws, size_t ws_size,
                              hipStream_t stream) {
    const float* x       = (const float*)d_in[0];   // [2048, 512]
    const int*   indices = (const int*)d_in[1];     // [2048]
    const float* weight  = (const float*)d_in[2];   // [64, 512, 512]
    const float* bias    = (const float*)d_in[3];   // [64, 512]
    float*       out     = (float*)d_out;           // [2048, 512]

    int* offsets = (int*)d_ws;                      // 65 ints
    int* ids     = offsets + 128;                   // 2048 ints (aligned block)

    sd_bucket_kernel<<<1, 256, 0, stream>>>(indices, offsets, ids);

    dim3 grid(NUM_OPTIONS * 4);
    sd_gemm_kernel<<<grid, 256, 0, stream>>>(x, weight, bias, offsets, ids, out);
}